// RadiusSearch_49581102465310
// MI455X (gfx1250) — compile-verified
//
#include <hip/hip_runtime.h>

#define K_MAX 128
#define CHUNK 4096  // points per LDS stage: 3 * 4096 * 4B = 48KB -> multiple blocks/WGP

typedef __attribute__((ext_vector_type(2))) float v2f;
typedef __attribute__((ext_vector_type(8))) float v8f;

// One wave = 16 queries (WMMA M-tile). One block = 8 waves = 128 queries.
// Points streamed through a 48KB LDS chunk (SoA). Per 16-point tile:
// V_WMMA_F32_16X16X4_F32 computes the 16x16 Q.P^T dot-product tile (K=3 padded
// to 4) on the matrix pipe; VALU does d2 = |q|^2+|p|^2-2*dot and ballot-based
// stream compaction into the padded [M, K_MAX] outputs. Two tiles per loop
// iteration so LDS loads batch ahead of one wait and 2 WMMAs overlap the
// previous tile's compaction.
__global__ __launch_bounds__(256) void radius_search_kernel(
    const float* __restrict__ points,   // [N,3]
    const float* __restrict__ queries,  // [M,3]
    const float* __restrict__ radii,    // [M]
    int*   __restrict__ out_idx,        // [M,K_MAX]
    float* __restrict__ out_dist,       // [M,K_MAX]
    int*   __restrict__ counts,         // [M]
    int N)
{
  __shared__ float sX[CHUNK];
  __shared__ float sY[CHUNK];
  __shared__ float sZ[CHUNK];

  const int tid    = threadIdx.x;
  const int lane   = tid & 31;
  const int lb     = lane & 15;          // lane-in-half == point column in tile
  const int sh     = lane & 16;          // half-select shift for ballot
  const unsigned prefmask = (1u << lb) - 1u;
  const int waveId = tid >> 5;
  const int qblock = blockIdx.x * 128;       // queries handled by this block
  const int qbase  = qblock + waveId * 16;   // queries handled by this wave

  // ---- pad-init this block's output rows ----
  for (int i = tid; i < 128 * K_MAX; i += blockDim.x) {
    out_idx [(size_t)qblock * K_MAX + i] = -1;
    out_dist[(size_t)qblock * K_MAX + i] = 0.0f;
  }

  // ---- per-lane query state ----
  // D-matrix layout: VGPR r holds row M=r in lanes 0-15, row M=8+r in lanes 16-31.
  const int h = (lane >= 16) ? 8 : 0;
  float qn[8], rr[8];
  int cnt[8];
  unsigned qoff[8];
#pragma unroll
  for (int r = 0; r < 8; ++r) {
    int q = qbase + r + h;
    float qx = queries[3 * q + 0];
    float qy = queries[3 * q + 1];
    float qz = queries[3 * q + 2];
    qn[r]   = qx * qx + qy * qy + qz * qz;
    float rad = radii[q];
    rr[r]   = rad * rad;
    cnt[r]  = 0;
    qoff[r] = (unsigned)q * K_MAX;  // 32-bit element offset for GVS-form stores
  }

  // A-matrix 16x4 f32: lanes 0-15 hold K={0,1}=(x,y), lanes 16-31 hold K={2,3}=(z,0)
  // for query rows M = lane&15.
  const int qa = qbase + lb;
  v2f A;
  if (lane < 16) { A.x = queries[3 * qa + 0]; A.y = queries[3 * qa + 1]; }
  else           { A.x = queries[3 * qa + 2]; A.y = 0.0f; }

  const float3* __restrict__ p3 = reinterpret_cast<const float3*>(points);

  // processes one 16-point tile whose lane-column point coords are (px,py,pz)
  auto do_tile = [&](int pIdx, float px, float py, float pz) {
    const float pn = px * px + py * py + pz * pz;
    v2f B;
    B.x = (lane < 16) ? px : pz;
    B.y = (lane < 16) ? py : 0.0f;
    v8f C = {};
    C = __builtin_amdgcn_wmma_f32_16x16x4_f32(
        /*neg_a=*/false, A, /*neg_b=*/false, B,
        /*c_mod=*/(short)0, C, /*reuse_a=*/false, /*reuse_b=*/false);
#pragma unroll
    for (int r = 0; r < 8; ++r) {
      float d2 = __builtin_fmaf(-2.0f, C[r], qn[r] + pn);
      bool  m  = d2 <= rr[r];
      unsigned bal  = __builtin_amdgcn_ballot_w32(m);
      unsigned half = bal >> sh;  // this lane's half: bits 0-15 = its query row
      if (m) {
        int slot = cnt[r] + __popc(half & prefmask);
        if (slot < K_MAX) {
          unsigned off = qoff[r] + (unsigned)slot;
          out_idx[off]  = pIdx;
          out_dist[off] = __builtin_fmaxf(d2, 0.0f);
        }
      }
      cnt[r] += __popc(half & 0xFFFFu);  // uncapped: row_splits needs exact counts
    }
  };

  // ---- stream points through LDS in 48KB chunks, ascending order ----
  for (int c0 = 0; c0 < N; c0 += CHUNK) {
    __syncthreads();
    for (int i = tid; i < CHUNK; i += blockDim.x) {
      int g = c0 + i;
      if (g < N) {
        float3 v = p3[g];  // 12B packed load
        sX[i] = v.x; sY[i] = v.y; sZ[i] = v.z;
      }
    }
    __syncthreads();

    // two tiles per iteration: batch 6 LDS loads, 2 WMMAs per body
    for (int t = 0; t < CHUNK; t += 32) {
      const int iA = t + lb;
      const int iB = t + 16 + lb;
      const float ax = sX[iA], ay = sY[iA], az = sZ[iA];
      const float bx = sX[iB], by = sY[iB], bz = sZ[iB];
      do_tile(c0 + iA, ax, ay, az);
      do_tile(c0 + iB, bx, by, bz);
    }
  }

  // one lane per half-wave publishes the exact counts
  if (lb == 0) {
#pragma unroll
    for (int r = 0; r < 8; ++r) counts[qbase + r + h] = cnt[r];
  }
}

// Exclusive prefix sum of counts[M] -> row_splits[M+1]; single 1024-thread block,
// 16 elements per thread, Hillis-Steele scan of the 1024 partials in LDS.
__global__ __launch_bounds__(1024) void scan_kernel(
    const int* __restrict__ counts, int* __restrict__ row_splits, int M)
{
  __shared__ int part[1024];
  const int t = threadIdx.x;
  const int base = t * 16;
  int local[16];
  int s = 0;
#pragma unroll
  for (int i = 0; i < 16; ++i) {
    local[i] = s;
    int idx = base + i;
    s += (idx < M) ? counts[idx] : 0;
  }
  part[t] = s;
  __syncthreads();
  for (int off = 1; off < 1024; off <<= 1) {
    int v = (t >= off) ? part[t - off] : 0;
    __syncthreads();
    part[t] += v;
    __syncthreads();
  }
  int prev = (t == 0) ? 0 : part[t - 1];
#pragma unroll
  for (int i = 0; i < 16; ++i) {
    int idx = base + i;
    if (idx < M) row_splits[idx] = prev + local[i];
  }
  if (t == 1023) row_splits[M] = part[1023];
}

extern "C" void kernel_launch(void* const* d_in, const int* in_sizes, int n_in,
                              void* d_out, int out_size, void* d_ws, size_t ws_size,
                              hipStream_t stream) {
  const float* points  = (const float*)d_in[0];  // [N,3]
  const float* queries = (const float*)d_in[1];  // [M,3]
  const float* radii   = (const float*)d_in[2];  // [M]
  const int N = in_sizes[0] / 3;
  const int M = in_sizes[2];

  // d_out flat layout (reference return order):
  //   [M*K_MAX] neighbors_index (int32) | [M+1] row_splits (int32) | [M*K_MAX] dist (f32)
  int*   out_idx    = (int*)d_out;
  int*   row_splits = (int*)d_out + (size_t)M * K_MAX;
  float* out_dist   = (float*)d_out + (size_t)M * K_MAX + (size_t)M + 1;
  int*   counts     = (int*)d_ws;

  const int blocks = M / 128;  // 128 queries per block
  radius_search_kernel<<<blocks, 256, 0, stream>>>(
      points, queries, radii, out_idx, out_dist, counts, N);
  scan_kernel<<<1, 1024, 0, stream>>>(counts, row_splits, M);
}